// ElmanRNN_60859686584479
// MI455X (gfx1250) — compile-verified
//
#include <hip/hip_runtime.h>
#include <math.h>

// Elman RNN on gfx1250 (MI455X), fp32 end-to-end using V_WMMA_F32_16X16X4_F32.
// BS=64, SEQ=512, INP=512, HID=1024.

#define BS   64
#define SEQ  512
#define INP  512
#define HID  1024

typedef __attribute__((ext_vector_type(2))) float v2f;
typedef __attribute__((ext_vector_type(8))) float v8f;

// ---------------------------------------------------------------------------
// WMMA fragment layout (V_WMMA_F32_16X16X4_F32, wave32):
//   A (16x4, MxK):  lane 0-15 -> M=lane, lane 16-31 -> M=lane-16
//                   VGPR j holds K = 2*(lane>=16) + j
//   B (4x16, KxN):  lane 0-15 -> N=lane, lane 16-31 -> N=lane-16
//                   VGPR j holds K = 2*(lane>=16) + j
//   C/D (16x16):    VGPR r -> M = r + 8*(lane>=16), N = lane&15
// ---------------------------------------------------------------------------

__device__ __forceinline__ v8f wmma_f32(v2f a, v2f b, v8f c) {
    // 8 args: (neg_a, A, neg_b, B, c_mod, C, reuse_a, reuse_b)
    return __builtin_amdgcn_wmma_f32_16x16x4_f32(false, a, false, b,
                                                 (short)0, c, false, false);
}

// ---------------------------------------------------------------------------
// Kernel 1: out[m, n] = sum_k x[m,k] * W_ih[n,k] + b_ih[n] + b_hh[n]
//   m = b*SEQ + s  (flat row, matches (B,S,H) output layout)
//   M = 32768, K = 512, N = 1024
// Block: 256 threads = 8 waves; wave w -> m-tile (blockIdx.x*8 + w).
// Each wave computes a 16(M) x 64(N) strip = 4 accumulator tiles.
// Grid: (32768/128, 1024/64) = (256, 16)
// ---------------------------------------------------------------------------
__global__ __launch_bounds__(256)
void elman_proj_kernel(const float* __restrict__ x,
                       const float* __restrict__ W_ih,
                       const float* __restrict__ b_ih,
                       const float* __restrict__ b_hh,
                       float* __restrict__ out) {
    const int tid  = threadIdx.x;
    const int wave = tid >> 5;
    const int lane = tid & 31;
    const int lo   = lane & 15;
    const int hi   = lane >> 4;        // 0 or 1
    const int koff = 2 * hi;           // K offset of this lane's fragment pair

    const int mbase  = (blockIdx.x * 8 + wave) * 16;
    const int nblock = blockIdx.y * 64;
    const int arow   = mbase + lo;     // A row this lane loads

    v8f acc0 = {}, acc1 = {}, acc2 = {}, acc3 = {};

    const float* xrow = x + (size_t)arow * INP;
    const float* w0 = W_ih + (size_t)(nblock +  0 + lo) * INP;
    const float* w1 = W_ih + (size_t)(nblock + 16 + lo) * INP;
    const float* w2 = W_ih + (size_t)(nblock + 32 + lo) * INP;
    const float* w3 = W_ih + (size_t)(nblock + 48 + lo) * INP;

    #pragma unroll 4
    for (int k = 0; k < INP; k += 4) {
        v2f a  = *(const v2f*)(xrow + k + koff);
        v2f f0 = *(const v2f*)(w0 + k + koff);
        v2f f1 = *(const v2f*)(w1 + k + koff);
        v2f f2 = *(const v2f*)(w2 + k + koff);
        v2f f3 = *(const v2f*)(w3 + k + koff);
        acc0 = wmma_f32(a, f0, acc0);
        acc1 = wmma_f32(a, f1, acc1);
        acc2 = wmma_f32(a, f2, acc2);
        acc3 = wmma_f32(a, f3, acc3);
    }

    // Epilogue: add biases, store. D layout: VGPR r -> M = mbase + r + 8*hi.
    v8f accs[4] = {acc0, acc1, acc2, acc3};
    #pragma unroll
    for (int t = 0; t < 4; ++t) {
        const int n = nblock + t * 16 + lo;
        const float bias = b_ih[n] + b_hh[n];
        #pragma unroll
        for (int r = 0; r < 8; ++r) {
            const int m = mbase + r + 8 * hi;
            out[(size_t)m * HID + n] = accs[t][r] + bias;
        }
    }
}

// ---------------------------------------------------------------------------
// Kernel 2: one recurrence step s:
//   out[b, s, h] = tanh(out[b, s, h] + sum_k out[b, s-1, k] * W_hh[h, k])
// (s==0: the h-term is zero). GEMM M=64, K=1024, N=1024, fused tanh.
//
// Block: 128 threads = 4 waves; each block owns ONE 16x16 output tile.
//   blockIdx.x -> n-tile (64), blockIdx.y -> m-tile (4)  => 256 blocks/step.
// K is split across the 4 waves (256 each); each wave uses 4 accumulators
// (16-deep dependent chains instead of 256) and partials are reduced via LDS.
// Wave 0 applies the fused tanh epilogue.
// ---------------------------------------------------------------------------
__global__ __launch_bounds__(128)
void elman_step_kernel(float* __restrict__ out,
                       const float* __restrict__ W_hh,
                       int s) {
    __shared__ float red[4][32][8];          // [wave][lane][acc-elem], 4 KB

    const int tid  = threadIdx.x;
    const int wave = tid >> 5;
    const int lane = tid & 31;
    const int lo   = lane & 15;
    const int hi   = lane >> 4;
    const int koff = 2 * hi;

    const int mbase = blockIdx.y * 16;       // batch rows of this tile
    const int nbase = blockIdx.x * 16;       // hidden cols of this tile
    const int arow  = mbase + lo;            // batch index for A loads
    const int n     = nbase + lo;            // hidden index for B loads

    if (s > 0) {
        const int kbase = wave * (HID / 4);  // this wave's K range [kbase, kbase+256)
        const float* hprev = out + (size_t)arow * (SEQ * HID)
                                 + (size_t)(s - 1) * HID + kbase;
        const float* wrow  = W_hh + (size_t)n * HID + kbase;

        v8f acc0 = {}, acc1 = {}, acc2 = {}, acc3 = {};
        for (int k = 0; k < HID / 4; k += 16) {
            v2f a0 = *(const v2f*)(hprev + k +  0 + koff);
            v2f a1 = *(const v2f*)(hprev + k +  4 + koff);
            v2f a2 = *(const v2f*)(hprev + k +  8 + koff);
            v2f a3 = *(const v2f*)(hprev + k + 12 + koff);
            v2f b0 = *(const v2f*)(wrow + k +  0 + koff);
            v2f b1 = *(const v2f*)(wrow + k +  4 + koff);
            v2f b2 = *(const v2f*)(wrow + k +  8 + koff);
            v2f b3 = *(const v2f*)(wrow + k + 12 + koff);
            acc0 = wmma_f32(a0, b0, acc0);
            acc1 = wmma_f32(a1, b1, acc1);
            acc2 = wmma_f32(a2, b2, acc2);
            acc3 = wmma_f32(a3, b3, acc3);
        }
        v8f part = acc0 + acc1 + acc2 + acc3;

        #pragma unroll
        for (int r = 0; r < 8; ++r)
            red[wave][lane][r] = part[r];
        __syncthreads();

        if (wave == 0) {
            #pragma unroll
            for (int r = 0; r < 8; ++r) {
                const float total = red[0][lane][r] + red[1][lane][r]
                                  + red[2][lane][r] + red[3][lane][r];
                const int m = mbase + r + 8 * hi;
                const size_t idx = (size_t)m * (SEQ * HID) + (size_t)s * HID + n;
                out[idx] = tanhf(out[idx] + total);
            }
        }
    } else {
        // s == 0: h_prev is zero -> just tanh the precomputed projection.
        if (wave == 0) {
            #pragma unroll
            for (int r = 0; r < 8; ++r) {
                const int m = mbase + r + 8 * hi;
                const size_t idx = (size_t)m * (SEQ * HID) + n;
                out[idx] = tanhf(out[idx]);
            }
        }
    }
}

// ---------------------------------------------------------------------------
// Launch: one projection GEMM, then 512 sequential step kernels (graph-friendly).
// ---------------------------------------------------------------------------
extern "C" void kernel_launch(void* const* d_in, const int* in_sizes, int n_in,
                              void* d_out, int out_size, void* d_ws, size_t ws_size,
                              hipStream_t stream) {
    const float* x    = (const float*)d_in[0];  // (BS, SEQ, INP)
    const float* W_ih = (const float*)d_in[1];  // (HID, INP)
    const float* W_hh = (const float*)d_in[2];  // (HID, HID)
    const float* b_ih = (const float*)d_in[3];  // (HID)
    const float* b_hh = (const float*)d_in[4];  // (HID)
    float* out = (float*)d_out;                 // (BS, SEQ, HID)

    dim3 projGrid((BS * SEQ) / 128, HID / 64);  // (256, 16)
    elman_proj_kernel<<<projGrid, 256, 0, stream>>>(x, W_ih, b_ih, b_hh, out);

    dim3 stepGrid(HID / 16, BS / 16);           // (64, 4) = 256 blocks/step
    for (int s = 0; s < SEQ; ++s) {
        elman_step_kernel<<<stepGrid, 128, 0, stream>>>(out, W_hh, s);
    }
}